// MultiStepLIFNode_20882130993593
// MI455X (gfx1250) — compile-verified
//
#include <hip/hip_runtime.h>
#include <cstdint>

// MultiStepLIFNode forward:
//   mem_t = 0.5*mem_{t-1} + 0.5*x_t ; s_t = (mem_t > 1.0) ; mem_t -= s_t
// x_seq: (T=4, 16, 256, 32, 32) fp32  ->  spike_seq same shape.
// Memory-bound (128 MB traffic, ~0.1 flop/byte). Strategy: per-thread float4,
// all 4 timestep tiles prefetched via CDNA5 async DMA to LDS (ASYNCcnt),
// drained in-order with s_wait_asynccnt, LIF state kept in VGPRs.

#define TSTEPS 4
#define BLOCK  256

__device__ __forceinline__ float lif_step(float& m, float x) {
    m = 0.5f * m + 0.5f * x;          // leak + input
    float s = (m > 1.0f) ? 1.0f : 0.0f; // heaviside(mem - v_th)
    m -= s;                            // soft reset
    return s;
}

__device__ __forceinline__ float4 lif_step4(float4& m, float4 x) {
    float4 s;
    s.x = lif_step(m.x, x.x);
    s.y = lif_step(m.y, x.y);
    s.z = lif_step(m.z, x.z);
    s.w = lif_step(m.w, x.w);
    return s;
}

__global__ __launch_bounds__(BLOCK) void lif_scan_kernel(
    const float4* __restrict__ xv,   // input, T*nvec float4s
    float4* __restrict__ ov,         // output, T*nvec float4s
    int nvec)                        // float4 elements per timestep
{
    __shared__ float4 smem[TSTEPS][BLOCK];

    const int tid = threadIdx.x;
    const int gid = blockIdx.x * BLOCK + tid;
    if (gid >= nvec) return;  // exact division in practice; safety guard

    // Global addresses for this lane's element at each timestep
    // (strided by 16 MB -> cannot fold into the 24-bit ioffset field).
    const float4* g0 = xv + gid;
    const float4* g1 = g0 + nvec;
    const float4* g2 = g1 + nvec;
    const float4* g3 = g2 + nvec;

    // LDS byte address of this lane's staging slot for t=0.
    // Generic pointer to __shared__: low 32 bits are the LDS offset.
    unsigned l0 = (unsigned)(uintptr_t)&smem[0][tid];
    unsigned l1 = l0 + 1u * BLOCK * 16u;
    unsigned l2 = l0 + 2u * BLOCK * 16u;
    unsigned l3 = l0 + 3u * BLOCK * 16u;

    // Fire-and-forget DMA of all 4 timesteps into LDS (ASYNCcnt += 4).
    // Async loads complete ("done") in issue order, so decreasing
    // s_wait_asynccnt values drain the pipeline one stage at a time.
    asm volatile("global_load_async_to_lds_b128 %0, %1, off"
                 :: "v"(l0), "v"(g0) : "memory");
    asm volatile("global_load_async_to_lds_b128 %0, %1, off"
                 :: "v"(l1), "v"(g1) : "memory");
    asm volatile("global_load_async_to_lds_b128 %0, %1, off"
                 :: "v"(l2), "v"(g2) : "memory");
    asm volatile("global_load_async_to_lds_b128 %0, %1, off"
                 :: "v"(l3), "v"(g3) : "memory");

    float4 m = make_float4(0.0f, 0.0f, 0.0f, 0.0f);
    float4 s;

    // t = 0
    asm volatile("s_wait_asynccnt 0x3" ::: "memory");
    s = lif_step4(m, smem[0][tid]);
    ov[gid] = s;

    // t = 1
    asm volatile("s_wait_asynccnt 0x2" ::: "memory");
    s = lif_step4(m, smem[1][tid]);
    ov[gid + nvec] = s;

    // t = 2
    asm volatile("s_wait_asynccnt 0x1" ::: "memory");
    s = lif_step4(m, smem[2][tid]);
    ov[gid + 2 * nvec] = s;

    // t = 3
    asm volatile("s_wait_asynccnt 0x0" ::: "memory");
    s = lif_step4(m, smem[3][tid]);
    ov[gid + 3 * nvec] = s;
}

extern "C" void kernel_launch(void* const* d_in, const int* in_sizes, int n_in,
                              void* d_out, int out_size, void* d_ws, size_t ws_size,
                              hipStream_t stream) {
    (void)n_in; (void)out_size; (void)d_ws; (void)ws_size;

    const float* x = (const float*)d_in[0];
    float* out = (float*)d_out;

    const long long total = (long long)in_sizes[0];     // T * 16*256*32*32
    const int per_t = (int)(total / TSTEPS);            // elements per timestep
    const int nvec  = per_t / 4;                        // float4s per timestep
    const int blocks = (nvec + BLOCK - 1) / BLOCK;      // 4096 for ref shape

    lif_scan_kernel<<<blocks, BLOCK, 0, stream>>>(
        reinterpret_cast<const float4*>(x),
        reinterpret_cast<float4*>(out),
        nvec);
}